// ClassQueryHeadPooling_45887430590916
// MI455X (gfx1250) — compile-verified
//
#include <hip/hip_runtime.h>

typedef float v2f __attribute__((ext_vector_type(2)));
typedef float v8f __attribute__((ext_vector_type(8)));
typedef unsigned int u32x4 __attribute__((ext_vector_type(4)));
typedef int i32x4 __attribute__((ext_vector_type(4)));
typedef int i32x8 __attribute__((ext_vector_type(8)));

#define NB   256      // graphs
#define NS   2048     // nodes per graph
#define ND   128      // hidden dim
#define NC   12       // classes
#define CP   16       // padded classes
#define CHUNK 128     // nodes staged per LDS tile
#define NCHUNK (NS / CHUNK)
#define QS   132      // Q row stride (pad 4 -> bank step 4)
#define SBS  2052     // score row stride
#define NTS  132      // node tile row stride (matches TDM pad: +4 dwords / 128 dwords)
#define AGS  132      // agg row stride
#define QSCALE 0.08838834764831845f  // 1/sqrt(128)

// Issue one TDM DMA: 128x128 f32 tile, global -> LDS, with 4-dword padding
// inserted every 128 dwords so rows land at stride NTS=132 (bank-conflict free).
// D# layout per CDNA5 ISA.
__device__ __forceinline__ void tdm_load_tile(const float* gsrc, float* ldsdst) {
  unsigned long long ga = (unsigned long long)gsrc;
  unsigned lds = (unsigned)(unsigned long long)ldsdst;   // LDS byte offset
  u32x4 g0;
  g0[0] = 1u;                                        // count=1, user mode
  g0[1] = lds;                                       // lds_addr
  g0[2] = (unsigned)(ga & 0xFFFFFFFFu);              // global_addr[31:0]
  g0[3] = (unsigned)((ga >> 32) & 0x1FFFFFFu) | (2u << 30);  // addr[56:32] | type=2
  i32x8 g1;
  g1[0] = (2 << 16)      // data_size = 4 bytes
        | (1 << 20)      // pad_enable
        | (6 << 22)      // pad_interval code 6 -> every 128 dwords (512B)
        | (3 << 25);     // pad_amount  code 3 -> 4 dwords (16B)
  g1[1] = (ND & 0xFFFF) << 16;                 // tensor_dim0[15:0]   (bits 63:48)
  g1[2] = ((CHUNK & 0xFFFF) << 16);            // dim0 hi=0 | tensor_dim1[15:0]
  g1[3] = (ND << 16);                          // dim1 hi=0 | tile_dim0 = 128
  g1[4] = CHUNK;                               // tile_dim1 = 128, tile_dim2 = 0
  g1[5] = ND;                                  // tensor_dim0_stride = 128 elements
  g1[6] = 0;                                   // stride hi | tensor_dim1_stride lo
  g1[7] = 0;
  i32x4 gz4 = {0, 0, 0, 0};
  i32x8 gz8 = {0, 0, 0, 0, 0, 0, 0, 0};
  // clang-23 / therock-10.0 lane: 6-arg form (g0, g1, g2, g3, g4, cpol)
  __builtin_amdgcn_tensor_load_to_lds(g0, g1, gz4, gz4, gz8, 0);
}

__global__ __launch_bounds__(256) void cqhp_kernel(
    const float* __restrict__ nodes, const float* __restrict__ queries,
    const float* __restrict__ gamma, const float* __restrict__ beta,
    const float* __restrict__ W1, const float* __restrict__ b1,
    const float* __restrict__ W2, const float* __restrict__ b2,
    float* __restrict__ logits, float* __restrict__ attn)
{
  extern __shared__ float smem[];
  float* Q   = smem;                  // CP x QS
  float* SB  = Q   + CP * QS;         // CP x SBS (whole per-graph score matrix)
  float* NT0 = SB  + CP * SBS;        // CHUNK x NTS (double-buffered node tiles)
  float* NT1 = NT0 + CHUNK * NTS;
  float* AG  = NT1 + CHUNK * NTS;     // CP x AGS (aggregated)

  const int tid  = threadIdx.x;
  const int lane = tid & 31;
  const int wid  = tid >> 5;          // 8 waves
  const int r    = lane & 15;
  const int h    = lane >> 4;         // K-half select per WMMA layout
  const int bid  = blockIdx.x;

  const float* gbase = nodes + (long)bid * NS * ND;

  // Kick off DMA of chunk 0 while we load the queries.
  if (wid == 0) tdm_load_tile(gbase, NT0);

  // Warm L2 with the MLP weights used in phase 4 (reused by all 256 blocks).
  __builtin_prefetch(W1 + tid * 32, 0, 0);

  // Load class queries (scaled by 1/sqrt(D)), zero-pad rows >= NC.
  for (int i = 0; i < 8; ++i) {
    int e = i * 256 + tid;
    int row = e >> 7, col = e & 127;
    Q[row * QS + col] = (row < NC) ? queries[row * ND + col] * QSCALE : 0.0f;
  }

  // ---------------- Phase 1: scores = Q * N^T  ->  SB ----------------
  for (int chunk = 0; chunk < NCHUNK; ++chunk) {
    float* cur = (chunk & 1) ? NT1 : NT0;
    float* nxt = (chunk & 1) ? NT0 : NT1;
    if (wid == 0) {
      if (chunk + 1 < NCHUNK) {
        tdm_load_tile(gbase + (long)(chunk + 1) * CHUNK * ND, nxt);
        __builtin_amdgcn_s_wait_tensorcnt(1);   // chunk's tile resident
      } else {
        __builtin_amdgcn_s_wait_tensorcnt(0);
      }
    }
    __syncthreads();

    v8f acc;
    #pragma unroll
    for (int i = 0; i < 8; ++i) acc[i] = 0.0f;
    const float* qrow = Q   + r * QS + 2 * h;
    const float* nrow = cur + (wid * 16 + r) * NTS + 2 * h;
    #pragma unroll
    for (int k0 = 0; k0 < ND; k0 += 4) {
      v2f a = *(const v2f*)(qrow + k0);   // A[16x4]: lane(r,h) -> A[r][k0+2h..+1]
      v2f b = *(const v2f*)(nrow + k0);   // B[4x16]=N^T: lane -> nodes[s0+r][k0+2h..+1]
      acc = __builtin_amdgcn_wmma_f32_16x16x4_f32(false, a, false, b,
                                                  (short)0, acc, false, false);
    }
    int col = chunk * CHUNK + wid * 16 + r;
    #pragma unroll
    for (int v = 0; v < 8; ++v)
      SB[(v + 8 * h) * SBS + col] = acc[v];   // D[m][n]: m=v+8h, n=r
    __syncthreads();
  }

  // ---------------- Phase 2: softmax rows, emit attn ----------------
  for (int cc = 0; cc < 2; ++cc) {
    int c = wid + 8 * cc;
    if (c < NC) {
      float* row = SB + c * SBS;
      float m = -INFINITY;
      for (int i = lane; i < NS; i += 32) m = fmaxf(m, row[i]);
      #pragma unroll
      for (int off = 16; off >= 1; off >>= 1) m = fmaxf(m, __shfl_xor(m, off, 32));
      float sum = 0.0f;
      for (int i = lane; i < NS; i += 32) {
        float e = __expf(row[i] - m); row[i] = e; sum += e;
      }
      #pragma unroll
      for (int off = 16; off >= 1; off >>= 1) sum += __shfl_xor(sum, off, 32);
      float inv = 1.0f / sum;
      float* aout = attn + ((long)bid * NC + c) * NS;
      for (int i = lane; i < NS; i += 32) {
        float p = row[i] * inv; row[i] = p; aout[i] = p;
      }
    }
  }
  __syncthreads();

  // ---------------- Phase 3: aggregated = P * N ----------------
  v8f agg;
  #pragma unroll
  for (int i = 0; i < 8; ++i) agg[i] = 0.0f;
  const int d0 = wid * 16;                  // this wave's feature tile
  if (wid == 0) tdm_load_tile(gbase, NT0);  // restart the stream
  for (int chunk = 0; chunk < NCHUNK; ++chunk) {
    float* cur = (chunk & 1) ? NT1 : NT0;
    float* nxt = (chunk & 1) ? NT0 : NT1;
    if (wid == 0) {
      if (chunk + 1 < NCHUNK) {
        tdm_load_tile(gbase + (long)(chunk + 1) * CHUNK * ND, nxt);
        __builtin_amdgcn_s_wait_tensorcnt(1);
      } else {
        __builtin_amdgcn_s_wait_tensorcnt(0);
      }
    }
    __syncthreads();

    const float* prow = SB + r * SBS + chunk * CHUNK + 2 * h;
    #pragma unroll
    for (int k0 = 0; k0 < CHUNK; k0 += 4) {
      v2f a = *(const v2f*)(prow + k0);     // A = P[r][s0+2h..+1]
      v2f b;
      b[0] = cur[(k0 + 2 * h)     * NTS + d0 + r];   // B[k][n] = nodes[s][d0+n]
      b[1] = cur[(k0 + 2 * h + 1) * NTS + d0 + r];
      agg = __builtin_amdgcn_wmma_f32_16x16x4_f32(false, a, false, b,
                                                  (short)0, agg, false, false);
    }
    __syncthreads();
  }
  #pragma unroll
  for (int v = 0; v < 8; ++v)
    AG[(v + 8 * h) * AGS + d0 + r] = agg[v];
  __syncthreads();

  // ---------------- Phase 4: LayerNorm + MLP head ----------------
  for (int cc = 0; cc < 2; ++cc) {
    int c = wid + 8 * cc;
    bool active = (c < NC);
    if (active) {
      float x[4]; float s = 0.0f;
      #pragma unroll
      for (int j = 0; j < 4; ++j) { x[j] = AG[c * AGS + lane + 32 * j]; s += x[j]; }
      #pragma unroll
      for (int off = 16; off >= 1; off >>= 1) s += __shfl_xor(s, off, 32);
      float mu = s * (1.0f / 128.0f);
      float vs = 0.0f;
      #pragma unroll
      for (int j = 0; j < 4; ++j) { float d = x[j] - mu; vs += d * d; }
      #pragma unroll
      for (int off = 16; off >= 1; off >>= 1) vs += __shfl_xor(vs, off, 32);
      float rs = rsqrtf(vs * (1.0f / 128.0f) + 1e-5f);
      #pragma unroll
      for (int j = 0; j < 4; ++j) {
        int idx = lane + 32 * j;
        AG[c * AGS + idx] = (x[j] - mu) * rs * gamma[idx] + beta[idx];
      }
    }
    __syncthreads();
    if (active) {
      int j0 = 2 * lane, j1 = j0 + 1;
      float h0 = b1[j0], h1 = b1[j1];
      for (int d = 0; d < ND; ++d) {
        float hd = AG[c * AGS + d];
        h0 = fmaf(hd, W1[d * 64 + j0], h0);
        h1 = fmaf(hd, W1[d * 64 + j1], h1);
      }
      h0 = fmaxf(h0, 0.0f); h1 = fmaxf(h1, 0.0f);
      float part = h0 * W2[j0] + h1 * W2[j1];
      #pragma unroll
      for (int off = 16; off >= 1; off >>= 1) part += __shfl_xor(part, off, 32);
      if (lane == 0) logits[bid * NC + c] = part + b2[0];
    }
    __syncthreads();
  }
}

extern "C" void kernel_launch(void* const* d_in, const int* in_sizes, int n_in,
                              void* d_out, int out_size, void* d_ws, size_t ws_size,
                              hipStream_t stream) {
  const float* nodes   = (const float*)d_in[0];
  // d_in[1] = batch (int64, sorted, equal-sized graphs) -> not needed
  const float* queries = (const float*)d_in[2];
  const float* gamma   = (const float*)d_in[3];
  const float* beta    = (const float*)d_in[4];
  const float* W1      = (const float*)d_in[5];
  const float* b1      = (const float*)d_in[6];
  const float* W2      = (const float*)d_in[7];
  const float* b2      = (const float*)d_in[8];

  float* logits = (float*)d_out;                       // [B, C]
  float* attn   = (float*)d_out + (size_t)NB * NC;     // [B, C, S]

  size_t smem = (size_t)(CP * QS + CP * SBS + 2 * CHUNK * NTS + CP * AGS) * sizeof(float);
  (void)hipFuncSetAttribute((const void*)cqhp_kernel,
                            hipFuncAttributeMaxDynamicSharedMemorySize, (int)smem);
  cqhp_kernel<<<NB, 256, smem, stream>>>(nodes, queries, gamma, beta,
                                         W1, b1, W2, b2, logits, attn);
}